// SSDBBox_51522427683150
// MI455X (gfx1250) — compile-verified
//
#include <hip/hip_runtime.h>
#include <float.h>
#include <math.h>

namespace {

constexpr int NLEV = 6;
constexpr int MTOT = 5480;          // 1000*5 + 480
constexpr int NBATCH = 64;
constexpr float F_SCORE_THR = 0.02f;
constexpr float F_IOU_THR   = 0.45f;
constexpr float F_MAX_SHAPE = 320.0f;
constexpr float F_MAX_RATIO = 4.1351665567423563f;   // |ln 0.016|

__constant__ int    c_HW[NLEV]   = {400,100,25,9,4,1};
__constant__ int    c_W[NLEV]    = {20,10,5,3,2,1};
__constant__ int    c_K[NLEV]    = {1000,1000,1000,1000,1000,480};
__constant__ int    c_LOFF[NLEV] = {0,1000,2000,3000,4000,5000};
__constant__ double c_ST[NLEV]   = {16.0,32.0,64.0,107.0,160.0,320.0};
__constant__ double c_MN[NLEV]   = {48.0,100.0,150.0,202.0,253.0,304.0};
__constant__ double c_MX[NLEV]   = {100.0,150.0,202.0,253.0,304.0,320.0};

// per-image workspace layout (bytes)
constexpr int OFF_BOXES  = 0;                      // MTOT*4 f32
constexpr int OFF_SCORES = MTOT * 16;              // MTOT f32
constexpr int OFF_LABELS = OFF_SCORES + MTOT * 4;  // MTOT i32
constexpr int OFF_ORD    = OFF_LABELS + MTOT * 4;  // MTOT i32
constexpr int OFF_META   = OFF_ORD + MTOT * 4;     // small
constexpr size_t IMG_WS  = 153600;                 // 600*256, >= OFF_META+64

struct LevelPtrs { const float* cls[NLEV]; const float* box[NLEV]; };

// ---- force AS(1): global_load/global_store instead of flat_* -----------
// (FLAT ops tick LOADcnt *and* DScnt and arbitrate against the LDS pipe;
//  this kernel keeps the LDS pipe busy, so keep global traffic off it.)
#define GAS __attribute__((address_space(1)))
__device__ __forceinline__ float gldf(const float* p) { return *(const GAS float*)p; }
__device__ __forceinline__ int   gldi(const int*   p) { return *(const GAS int*)p; }
__device__ __forceinline__ void  gstf(float* p, float v) { *(GAS float*)p = v; }
__device__ __forceinline__ void  gsti(int*   p, int   v) { *(GAS int*)p = v; }

__device__ __forceinline__ void assume_global(const void* p) {
  void* q = const_cast<void*>(p);
  __builtin_assume(!__builtin_amdgcn_is_shared(q));
  __builtin_assume(!__builtin_amdgcn_is_private(q));
}

// ---- CDNA5 async global->LDS copy (ASYNCcnt path) ----------------------
__device__ __forceinline__ void async_copy_b32_to_lds(void* lds_dst, const void* gsrc) {
  unsigned lds_off = (unsigned)(unsigned long long)lds_dst; // low 32b of flat LDS ptr = LDS offset
  asm volatile("global_load_async_to_lds_b32 %0, %1, off"
               :: "v"(lds_off), "v"(gsrc) : "memory");
}
__device__ __forceinline__ void wait_asynccnt0() {
  asm volatile("s_wait_asynccnt 0x0" ::: "memory");
}

// Deterministic score recompute (noinline => identical bits across all passes)
__device__ __attribute__((noinline))
float score_of(const float* __restrict__ clsb, unsigned hw, unsigned i,
               const float* __restrict__ rowm, const float* __restrict__ rowinv) {
  unsigned r = i / 80u;
  unsigned c = i - r * 80u;
  unsigned pos = r / 6u;
  unsigned a = r - pos * 6u;
  float x = gldf(&clsb[(a * 81u + c) * hw + pos]);   // global_load_b32
  return expf(x - rowm[r]) * rowinv[r];
}

// ======================= kernel 1: softmax + exact top-K ================
__global__ __launch_bounds__(256)
void k_select(LevelPtrs P, char* __restrict__ ws) {
  const int lvl = blockIdx.x;
  const int img = blockIdx.y;
  const int tid = threadIdx.x;
  const unsigned hw = (unsigned)c_HW[lvl];
  const unsigned Wd = (unsigned)c_W[lvl];
  const int rows = (int)hw * 6;
  const unsigned N = (unsigned)rows * 80u;
  const unsigned K = (unsigned)c_K[lvl];

  const float* __restrict__ clsb = P.cls[lvl] + (size_t)img * 486u * hw;
  const float* __restrict__ boxb = P.box[lvl] + (size_t)img * 24u  * hw;
  assume_global(clsb);
  assume_global(boxb);
  assume_global(ws);

  char* wsb = ws + (size_t)img * IMG_WS;
  float* g_boxes  = (float*)(wsb + OFF_BOXES);
  float* g_scores = (float*)(wsb + OFF_SCORES);
  int*   g_labels = (int*)  (wsb + OFF_LABELS);

  __shared__ float    s_rowm[2400];
  __shared__ float    s_rowinv[2400];
  __shared__ float    s_delta[9600];
  __shared__ unsigned s_hist[256];
  __shared__ unsigned s_sc[4];

  // default-fill this level's K slots (poison safety)
  for (unsigned s = tid; s < K; s += 256) {
    unsigned gi = (unsigned)c_LOFF[lvl] + s;
    gstf(&g_boxes[gi*4+0], 0.f); gstf(&g_boxes[gi*4+1], 0.f);
    gstf(&g_boxes[gi*4+2], 0.f); gstf(&g_boxes[gi*4+3], 0.f);
    gstf(&g_scores[gi], -1.f);   gsti(&g_labels[gi], 0);
  }

  __builtin_prefetch(clsb + tid, 0, 0);   // global_prefetch_b8

  // phase 0: per-row softmax stats (max, 1/sum over 81 channels)
  for (int r = tid; r < rows; r += 256) {
    unsigned pos = (unsigned)r / 6u, a = (unsigned)r - pos * 6u;
    const float* p = clsb + (a * 81u) * hw + pos;
    float m = -FLT_MAX;
    for (int c = 0; c < 81; ++c) m = fmaxf(m, gldf(&p[(unsigned)c * hw]));
    float ssum = 0.f;
    for (int c = 0; c < 81; ++c) ssum += expf(gldf(&p[(unsigned)c * hw]) - m);
    s_rowm[r] = m; s_rowinv[r] = 1.0f / ssum;
  }
  // async-stage box deltas into LDS: s_delta[r*4+k]
  for (int i = tid; i < rows * 4; i += 256) {
    unsigned r = (unsigned)i >> 2, k4 = (unsigned)i & 3u;
    unsigned pos = r / 6u, a = r - pos * 6u;
    async_copy_b32_to_lds(&s_delta[i], boxb + (a * 4u + k4) * hw + pos);
  }
  wait_asynccnt0();
  __syncthreads();

  // phase 1: 4-pass MSB radix-select for exact K-th-largest key
  unsigned prefix = 0u, need = K;
  for (int pass = 0; pass < 4; ++pass) {
    const int shift = 24 - 8 * pass;
    s_hist[tid] = 0u;  // blockDim==256
    __syncthreads();
    const unsigned pmask = pass ? (0xFFFFFFFFu << (shift + 8)) : 0u;
    for (unsigned i = tid; i < N; i += 256) {
      unsigned key = __float_as_uint(score_of(clsb, hw, i, s_rowm, s_rowinv));
      if ((key & pmask) == prefix)
        atomicAdd(&s_hist[(key >> shift) & 255u], 1u);
    }
    __syncthreads();
    if (tid == 0) {
      unsigned acc = 0; int t = 0;
      for (int b = 255; b >= 0; --b) {
        unsigned h = s_hist[b];
        if (acc + h >= need) { t = b; break; }
        acc += h;
      }
      s_sc[0] = prefix | ((unsigned)t << shift);
      s_sc[1] = need - acc;
    }
    __syncthreads();
    prefix = s_sc[0]; need = s_sc[1];
    __syncthreads();
  }
  const unsigned T = prefix;          // exact threshold key
  const unsigned G = K - need;        // # strictly greater
  if (tid == 0) { s_sc[2] = 0u; s_sc[3] = 0u; }
  __syncthreads();

  // phase 2: compaction + anchor decode
  for (unsigned i = tid; i < N; i += 256) {
    float val = score_of(clsb, hw, i, s_rowm, s_rowinv);
    unsigned key = __float_as_uint(val);
    int slot = -1;
    if (key > T) slot = (int)atomicAdd(&s_sc[2], 1u);
    else if (key == T) {
      unsigned e = atomicAdd(&s_sc[3], 1u);
      if (e < need) slot = (int)(G + e);
    }
    if (slot < 0) continue;

    unsigned r = i / 80u, c = i - r * 80u;
    unsigned pos = r / 6u, a = r - pos * 6u;

    // analytic SSD anchor (float64, matching numpy precompute)
    double mn = c_MN[lvl], mx = c_MX[lvl], st = c_ST[lvl];
    double scale = (a == 1u) ? sqrt(mx / mn) : 1.0;
    double ratio = (a == 2u) ? 0.5 : (a == 3u) ? 2.0
                 : (a == 4u) ? (1.0/3.0) : (a == 5u) ? 3.0 : 1.0;
    double sr = sqrt(ratio);
    double wa = mn * scale / sr, ha = mn * scale * sr;
    double cx = 0.5 * st + (double)(pos % Wd) * st;
    double cy = 0.5 * st + (double)(pos / Wd) * st;
    float x1 = (float)(cx - 0.5 * wa), y1 = (float)(cy - 0.5 * ha);
    float x2 = (float)(cx + 0.5 * wa), y2 = (float)(cy + 0.5 * ha);

    // delta2bbox
    float d0 = s_delta[r*4+0] * 0.1f;
    float d1 = s_delta[r*4+1] * 0.1f;
    float d2 = fminf(fmaxf(s_delta[r*4+2] * 0.2f, -F_MAX_RATIO), F_MAX_RATIO);
    float d3 = fminf(fmaxf(s_delta[r*4+3] * 0.2f, -F_MAX_RATIO), F_MAX_RATIO);
    float pxc = (x1 + x2) * 0.5f, pyc = (y1 + y2) * 0.5f;
    float pw = x2 - x1, ph = y2 - y1;
    float gx = pxc + pw * d0, gy = pyc + ph * d1;
    float gw = pw * expf(d2), gh = ph * expf(d3);
    float b0 = fminf(fmaxf(gx - 0.5f * gw, 0.f), F_MAX_SHAPE);
    float b1 = fminf(fmaxf(gy - 0.5f * gh, 0.f), F_MAX_SHAPE);
    float b2 = fminf(fmaxf(gx + 0.5f * gw, 0.f), F_MAX_SHAPE);
    float b3 = fminf(fmaxf(gy + 0.5f * gh, 0.f), F_MAX_SHAPE);

    bool valid = val > F_SCORE_THR;
    unsigned gi = (unsigned)c_LOFF[lvl] + (unsigned)slot;
    gstf(&g_boxes[gi*4+0], valid ? b0 : 0.f);
    gstf(&g_boxes[gi*4+1], valid ? b1 : 0.f);
    gstf(&g_boxes[gi*4+2], valid ? b2 : 0.f);
    gstf(&g_boxes[gi*4+3], valid ? b3 : 0.f);
    gstf(&g_scores[gi], valid ? val : -1.f);
    gsti(&g_labels[gi], (int)c);
  }
}

// ======================= kernel 2: per-image score sort =================
__global__ __launch_bounds__(1024)
void k_sort(char* __restrict__ ws) {
  const int img = blockIdx.x, tid = threadIdx.x;
  assume_global(ws);
  char* wsb = ws + (size_t)img * IMG_WS;
  const float* g_scores = (const float*)(wsb + OFF_SCORES);
  const float* g_boxes  = (const float*)(wsb + OFF_BOXES);
  int*   g_ord  = (int*)  (wsb + OFF_ORD);
  float* g_meta = (float*)(wsb + OFF_META);

  __shared__ float ss[8192];
  __shared__ int   si[8192];
  __shared__ int   s_max;

  if (tid == 0) s_max = 0;   // boxes >= 0 always
  for (int m = tid; m < 8192; m += 1024) {
    if (m < MTOT) { ss[m] = gldf(&g_scores[m]); si[m] = m; }
    else          { ss[m] = -FLT_MAX;           si[m] = m; }
  }
  int lm = 0;
  for (int q = tid; q < MTOT * 4; q += 1024) lm = max(lm, __float_as_int(gldf(&g_boxes[q])));
  atomicMax(&s_max, lm);

  // bitonic sort, descending by score
  for (int k = 2; k <= 8192; k <<= 1) {
    for (int j = k >> 1; j > 0; j >>= 1) {
      __syncthreads();
      for (int idx = tid; idx < 8192; idx += 1024) {
        int ixj = idx ^ j;
        if (ixj > idx) {
          bool desc = ((idx & k) == 0);
          float a = ss[idx], b = ss[ixj];
          if (desc ? (a < b) : (a > b)) {
            ss[idx] = b; ss[ixj] = a;
            int t = si[idx]; si[idx] = si[ixj]; si[ixj] = t;
          }
        }
      }
    }
  }
  __syncthreads();
  for (int m = tid; m < MTOT; m += 1024) gsti(&g_ord[m], si[m]);
  if (tid == 0) gstf(&g_meta[0], __int_as_float(s_max));
}

// ======================= kernel 3: greedy NMS + emit ====================
__global__ __launch_bounds__(1024)
void k_nms(char* __restrict__ ws, float* __restrict__ out) {
  const int img = blockIdx.x, tid = threadIdx.x;
  assume_global(ws);
  assume_global(out);
  char* wsb = ws + (size_t)img * IMG_WS;
  const float* g_boxes  = (const float*)(wsb + OFF_BOXES);
  const float* g_scores = (const float*)(wsb + OFF_SCORES);
  const int*   g_labels = (const int*)  (wsb + OFF_LABELS);
  const int*   g_ord    = (const int*)  (wsb + OFF_ORD);
  const float* g_meta   = (const float*)(wsb + OFF_META);

  __shared__ float sx0[MTOT], sy0[MTOT], sx1[MTOT], sy1[MTOT], sar[MTOT];
  __shared__ int   skp[MTOT];

  const float offs = gldf(&g_meta[0]) + 1.0f;
  for (int m = tid; m < MTOT; m += 1024) {
    int j = gldi(&g_ord[m]);
    float o = (float)gldi(&g_labels[j]) * offs;  // per-class offset trick
    float x0 = gldf(&g_boxes[j*4+0]) + o, y0 = gldf(&g_boxes[j*4+1]) + o;
    float x1 = gldf(&g_boxes[j*4+2]) + o, y1 = gldf(&g_boxes[j*4+3]) + o;
    sx0[m] = x0; sy0[m] = y0; sx1[m] = x1; sy1[m] = y1;
    sar[m] = (x1 - x0) * (y1 - y0);
    skp[m] = (gldf(&g_scores[j]) > F_SCORE_THR) ? 1 : 0;
  }

  int nsurv = 0;
  for (int i = 0; i < MTOT; ++i) {
    __syncthreads();
    if (skp[i]) {
      ++nsurv;
      if (nsurv >= 100) break;  // survivors past #100 never reach the output
      float xi0 = sx0[i], yi0 = sy0[i], xi1 = sx1[i], yi1 = sy1[i], ai = sar[i];
      for (int j = i + 1 + tid; j < MTOT; j += 1024) {
        if (!skp[j]) continue;
        float lx = fmaxf(xi0, sx0[j]), ly = fmaxf(yi0, sy0[j]);
        float rx = fminf(xi1, sx1[j]), ry = fminf(yi1, sy1[j]);
        float inter = fmaxf(rx - lx, 0.f) * fmaxf(ry - ly, 0.f);
        float den = ai + sar[j] - inter;
        float iou = (den > 0.f) ? (inter / den) : 0.f;
        if (iou > F_IOU_THR) skp[j] = 0;
      }
    }
  }
  __syncthreads();

  if (tid == 0) {
    float* dets = out + (size_t)img * 500;
    float* labs = out + (size_t)NBATCH * 500 + (size_t)img * 100;
    int cnt = 0;
    for (int m = 0; m < MTOT && cnt < 100; ++m) {
      if (!skp[m]) continue;
      int j = gldi(&g_ord[m]);
      gstf(&dets[cnt*5+0], gldf(&g_boxes[j*4+0]));
      gstf(&dets[cnt*5+1], gldf(&g_boxes[j*4+1]));
      gstf(&dets[cnt*5+2], gldf(&g_boxes[j*4+2]));
      gstf(&dets[cnt*5+3], gldf(&g_boxes[j*4+3]));
      gstf(&dets[cnt*5+4], gldf(&g_scores[j]));
      gstf(&labs[cnt], (float)gldi(&g_labels[j]));
      ++cnt;
    }
    for (; cnt < 100; ++cnt) {
      gstf(&dets[cnt*5+0], 0.f); gstf(&dets[cnt*5+1], 0.f); gstf(&dets[cnt*5+2], 0.f);
      gstf(&dets[cnt*5+3], 0.f); gstf(&dets[cnt*5+4], 0.f);
      gstf(&labs[cnt], -1.f);
    }
  }
}

} // anonymous namespace

extern "C" void kernel_launch(void* const* d_in, const int* in_sizes, int n_in,
                              void* d_out, int out_size, void* d_ws, size_t ws_size,
                              hipStream_t stream) {
  (void)in_sizes; (void)n_in; (void)out_size; (void)ws_size;
  LevelPtrs P;
  for (int l = 0; l < NLEV; ++l) {
    P.cls[l] = (const float*)d_in[2 * l + 0];  // setup_inputs order: cls0,box0,cls1,box1,...
    P.box[l] = (const float*)d_in[2 * l + 1];
  }
  char* ws = (char*)d_ws;  // needs 64 * 153600 B ~= 9.8 MB
  dim3 g1(NLEV, NBATCH);
  k_select<<<g1, 256, 0, stream>>>(P, ws);
  k_sort  <<<NBATCH, 1024, 0, stream>>>(ws);
  k_nms   <<<NBATCH, 1024, 0, stream>>>(ws, (float*)d_out);
}